// PCNO_7318624272558
// MI455X (gfx1250) — compile-verified
//
#include <hip/hip_runtime.h>

#define Bb 2
#define Cc 64
#define Nn 16384
#define Dd 3
#define Kk 128
#define Ee 262144
#define Gg 3

typedef float v2f __attribute__((ext_vector_type(2)));
typedef float v8f __attribute__((ext_vector_type(8)));
typedef unsigned int v4u __attribute__((ext_vector_type(4)));
typedef int v8i __attribute__((ext_vector_type(8)));
typedef int v4i __attribute__((ext_vector_type(4)));

static __device__ __forceinline__ v8f wmma4(v2f a, v2f b, v8f c) {
  // D = A(16x4 f32) * B(4x16 f32) + C(16x16 f32)
  return __builtin_amdgcn_wmma_f32_16x16x4_f32(false, a, false, b, (short)0, c,
                                               false, false);
}

// TDM: DMA a 2D tile (tile_w x tile_h, 4-byte elems, row stride in elems) from
// global into LDS, padding 1 DWORD after every 32 DWORDs (LDS pitch = tw+1).
static __device__ __forceinline__ void tdm_load_2d(const void* gptr,
                                                   unsigned lds_byte,
                                                   unsigned tw, unsigned th,
                                                   unsigned stride_elems) {
  unsigned long long ga = (unsigned long long)gptr;
  v4u g0;
  g0[0] = 1u;                                    // count=1, user descriptor
  g0[1] = lds_byte;                              // lds_addr (bytes)
  g0[2] = (unsigned)(ga & 0xFFFFFFFFu);          // global_addr[31:0]
  g0[3] = (unsigned)((ga >> 32) & 0x01FFFFFFu) | (2u << 30);  // addr[56:32]|type=2
  const unsigned td0 = stride_elems;             // tensor_dim0 (OOB bound)
  const unsigned td1 = th;                       // tensor_dim1
  const unsigned long long s0 = stride_elems;    // tensor_dim0_stride
  v8i g1;
  g1[0] = (int)((2u << 16)      // data_size = 4B
                | (1u << 20)    // pad_enable
                | (4u << 22)    // pad_interval code 4 = 32 DWORDs
                | (0u << 25));  // pad_amount code 0 = 1 DWORD
  g1[1] = (int)((td0 & 0xFFFFu) << 16);
  g1[2] = (int)(((td0 >> 16) & 0xFFFFu) | ((td1 & 0xFFFFu) << 16));
  g1[3] = (int)(((td1 >> 16) & 0xFFFFu) | ((tw & 0xFFFFu) << 16));  // tile_dim0
  g1[4] = (int)(th & 0xFFFFu);                   // tile_dim1 (tile_dim2 = 0)
  g1[5] = (int)(unsigned)(s0 & 0xFFFFFFFFu);
  g1[6] = (int)(unsigned)((s0 >> 32) & 0xFFFFu); // dim1_stride = 0
  g1[7] = 0;
  v4i g2 = {0, 0, 0, 0};
  v4i g3 = {0, 0, 0, 0};
  v8i g4 = {0, 0, 0, 0, 0, 0, 0, 0};             // extra group (clang-23 6-arg form)
  __builtin_amdgcn_tensor_load_to_lds(g0, g1, g2, g3, g4, 0);
}

// ---------------- workspace layout (float offsets) ----------------
#define OFF_XC   0                         // [B][C][K]  x @ (cos*nw)
#define OFF_XS   (OFF_XC + Bb*Cc*Kk)       // [B][C][K]  x @ (sin*nw)  (un-negated)
#define OFF_X0   (OFF_XS + Bb*Cc*Kk)       // [B][C]     x @ nw
#define OFF_X3   (OFF_X0 + Bb*Cc)          // [B][N][C]  edge-scatter accumulator
#define ZERO_CNT (OFF_X3 + Bb*Nn*Cc)
#define OFF_BC   ZERO_CNT                  // [B][K][N]  cos bases
#define OFF_BS   (OFF_BC + Bb*Kk*Nn)       // [B][K][N]  sin bases
#define OFF_FC2  (OFF_BS + Bb*Kk*Nn)       // [B][C][K]  2*f_c
#define OFF_FS2  (OFF_FC2 + Bb*Cc*Kk)      // [B][C][K]  -2*f_s
#define OFF_F0   (OFF_FS2 + Bb*Cc*Kk)      // [B][C]     f_0
#define OFF_Y    (OFF_F0 + Bb*Cc)          // [D][B][N][C]  y_d = gw_d @ x
#define OFF_H    (OFF_Y + Dd*Bb*Nn*Cc)     // [B][C][N]  softsign(geo_wx@geo)*(wx@x)

__global__ void zero_kernel(float* __restrict__ p, int n) {
  for (int i = blockIdx.x * blockDim.x + threadIdx.x; i < n;
       i += gridDim.x * blockDim.x)
    p[i] = 0.f;
}

__global__ void bases_kernel(const float* __restrict__ nodes,
                             const float* __restrict__ modes,
                             float* __restrict__ bC, float* __restrict__ bS) {
  int idx = blockIdx.x * blockDim.x + threadIdx.x;
  if (idx >= Bb * Nn) return;
  int b = idx / Nn, n = idx % Nn;
  float d0 = nodes[(b * Nn + n) * Dd + 0];
  float d1 = nodes[(b * Nn + n) * Dd + 1];
  float d2 = nodes[(b * Nn + n) * Dd + 2];
  for (int k = 0; k < Kk; ++k) {
    float t = d0 * modes[k * Dd + 0] + d1 * modes[k * Dd + 1] + d2 * modes[k * Dd + 2];
    float s, c;
    sincosf(t, &s, &c);
    bC[(b * Kk + k) * Nn + n] = c;
    bS[(b * Kk + k) * Nn + n] = s;
  }
}

// forward transform: xc[b,i,k] += sum_n x[b,i,n]*cos[b,k,n]*nw[b,n]  (and sin)
__global__ void forward_kernel(const float* __restrict__ x,
                               const float* __restrict__ nodew,
                               const float* __restrict__ bC,
                               const float* __restrict__ bS,
                               float* __restrict__ xc, float* __restrict__ xs,
                               float* __restrict__ x0) {
  const int b = blockIdx.y;
  const int nbase = blockIdx.x * 256;
  const int tid = threadIdx.x;
  const int wv = tid >> 5, lane = tid & 31;
  const int half = lane >> 4, l = lane & 15;
  const int i0 = (wv & 3) * 16;
  const int kg = (wv >> 2) * 4;
  const float* xb  = x + b * Cc * Nn;
  const float* nwb = nodew + b * Nn;
  const float* bCb = bC + b * Kk * Nn;
  const float* bSb = bS + b * Kk * Nn;

  const v8f vz = {0.f, 0.f, 0.f, 0.f, 0.f, 0.f, 0.f, 0.f};
  v8f accC[4] = {vz, vz, vz, vz};
  v8f accS[4] = {vz, vz, vz, vz};

  for (int n = nbase; n < nbase + 256; n += 4) {
    const int nr = n + 2 * half;
    v2f a;
    a.x = xb[(i0 + l) * Nn + nr];
    a.y = xb[(i0 + l) * Nn + nr + 1];
    const float w0 = nwb[nr], w1 = nwb[nr + 1];
#pragma unroll
    for (int t = 0; t < 4; ++t) {
      const int k0 = (kg + t) * 16;
      v2f bc, bs;
      bc.x = bCb[(k0 + l) * Nn + nr] * w0;
      bc.y = bCb[(k0 + l) * Nn + nr + 1] * w1;
      bs.x = bSb[(k0 + l) * Nn + nr] * w0;
      bs.y = bSb[(k0 + l) * Nn + nr + 1] * w1;
      accC[t] = wmma4(a, bc, accC[t]);
      accS[t] = wmma4(a, bs, accS[t]);
    }
  }
#pragma unroll
  for (int t = 0; t < 4; ++t) {
    const int k0 = (kg + t) * 16;
#pragma unroll
    for (int j = 0; j < 8; ++j) {
      const int ii = i0 + j + 8 * half;
      const int kk = k0 + l;
      atomicAdd(&xc[(b * Cc + ii) * Kk + kk], accC[t][j]);
      atomicAdd(&xs[(b * Cc + ii) * Kk + kk], accS[t][j]);
    }
  }
  if (tid < Cc) {
    float s = 0.f;
    for (int n = nbase; n < nbase + 256; ++n) s += xb[tid * Nn + n] * nwb[n];
    atomicAdd(&x0[b * Cc + tid], s);
  }
}

__global__ void middle_kernel(const float* __restrict__ xc, const float* __restrict__ xs,
                              const float* __restrict__ x0, const float* __restrict__ wc,
                              const float* __restrict__ wsn, const float* __restrict__ w0,
                              float* __restrict__ fc2, float* __restrict__ fs2,
                              float* __restrict__ f0) {
  const int b = blockIdx.x / Kk, k = blockIdx.x % Kk, o = threadIdx.x;
  float fc = 0.f, fs = 0.f;
  for (int i = 0; i < Cc; ++i) {
    const float xcv = xc[(b * Cc + i) * Kk + k];   // x_c
    const float xsv = xs[(b * Cc + i) * Kk + k];   // -x_s
    const float wcv = wc[(i * Cc + o) * Kk + k];
    const float wsv = wsn[(i * Cc + o) * Kk + k];
    fc += xcv * wcv + xsv * wsv;   // f_c
    fs += xcv * wsv - xsv * wcv;   // f_s
  }
  fc2[(b * Cc + o) * Kk + k] = 2.f * fc;
  fs2[(b * Cc + o) * Kk + k] = -2.f * fs;
  if (k == 0) {
    float f = 0.f;
    for (int i = 0; i < Cc; ++i) f += x0[b * Cc + i] * w0[i * Cc + o];
    f0[b * Cc + o] = f;
  }
}

__global__ void gemm64_kernel(const float* __restrict__ A, int ars, int acs, int aoff,
                              const float* __restrict__ X, float* __restrict__ Out,
                              int transOut) {
  const int b = blockIdx.y;
  const int tid = threadIdx.x;
  const int wv = tid >> 5, lane = tid & 31;
  const int half = lane >> 4, l = lane & 15;
  const int i0 = (wv & 3) * 16;
  const int n0 = blockIdx.x * 32 + (wv >> 2) * 16;
  const float* Xb = X + b * Cc * Nn;
  v8f acc = {0.f, 0.f, 0.f, 0.f, 0.f, 0.f, 0.f, 0.f};
#pragma unroll
  for (int c = 0; c < Cc; c += 4) {
    const int cr = c + 2 * half;
    v2f a, bf;
    a.x = A[(i0 + l) * ars + cr * acs + aoff];
    a.y = A[(i0 + l) * ars + (cr + 1) * acs + aoff];
    bf.x = Xb[cr * Nn + n0 + l];
    bf.y = Xb[(cr + 1) * Nn + n0 + l];
    acc = wmma4(a, bf, acc);
  }
#pragma unroll
  for (int j = 0; j < 8; ++j) {
    const int o = i0 + j + 8 * half, n = n0 + l;
    if (transOut) Out[(b * Nn + n) * Cc + o] = acc[j];
    else          Out[(b * Cc + o) * Nn + n] = acc[j];
  }
}

__global__ void h_kernel(const float* __restrict__ geo_wx,
                         const float* __restrict__ geo, float* __restrict__ h) {
  int idx = blockIdx.x * blockDim.x + threadIdx.x;
  if (idx >= Bb * Cc * Nn) return;
  const int n = idx % Nn;
  const int o = (idx / Nn) % Cc;
  const int b = idx / (Cc * Nn);
  float s = 0.f;
  for (int g = 0; g < Gg; ++g)
    s += geo_wx[o * Gg + g] * geo[(b * Gg + g) * Nn + n];
  h[idx] = (s / (1.f + fabsf(s))) * h[idx];
}

__global__ void edge_kernel(const int* __restrict__ edges, const float* __restrict__ egw,
                            const float* __restrict__ y, float* __restrict__ x3) {
  const int gt = blockIdx.x * blockDim.x + threadIdx.x;
  const int lane = gt & 31;
  const int we = gt >> 5;
  const int b = we / Ee, e = we % Ee;
  const int tgt = edges[(b * Ee + e) * 2 + 0];
  const int src = edges[(b * Ee + e) * 2 + 1];
  const float g0 = egw[(b * Ee + e) * Dd + 0];
  const float g1 = egw[(b * Ee + e) * Dd + 1];
  const float g2 = egw[(b * Ee + e) * Dd + 2];
#pragma unroll
  for (int p = 0; p < 2; ++p) {
    const int o = lane + 32 * p;
    float v = 0.f;
    v += g0 * (y[((0 * Bb + b) * Nn + src) * Cc + o] - y[((0 * Bb + b) * Nn + tgt) * Cc + o]);
    v += g1 * (y[((1 * Bb + b) * Nn + src) * Cc + o] - y[((1 * Bb + b) * Nn + tgt) * Cc + o]);
    v += g2 * (y[((2 * Bb + b) * Nn + src) * Cc + o] - y[((2 * Bb + b) * Nn + tgt) * Cc + o]);
    atomicAdd(&x3[(b * Nn + tgt) * Cc + o], v);
  }
}

// fused inverse transform + x2 + x4 with TDM-staged LDS tiles.
#define TW 32                       // tile width (n columns per block)
#define TP 33                       // padded LDS row pitch (floats)
#define LDS_BC 0
#define LDS_BS (LDS_BC + Kk * TP)
#define LDS_X  (LDS_BS + Kk * TP)
#define LDS_H  (LDS_X + Cc * TP)
#define LDS_TOT (LDS_H + Cc * TP)   // 12672 floats = 50688 B

__global__ void final_kernel(const float* __restrict__ fc2, const float* __restrict__ fs2,
                             const float* __restrict__ f0, const float* __restrict__ bC,
                             const float* __restrict__ bS, const float* __restrict__ W,
                             const float* __restrict__ x, const float* __restrict__ w2,
                             const float* __restrict__ h, const float* __restrict__ x3,
                             float* __restrict__ out) {
  __shared__ float smem[LDS_TOT];
  const int b = blockIdx.y;
  const int nblk = blockIdx.x * TW;
  const int tid = threadIdx.x;
  const int wv = tid >> 5, lane = tid & 31;
  const int half = lane >> 4, l = lane & 15;
  const int i0 = (wv & 3) * 16;
  const int col = (wv >> 2) * 16 + l;

  if (tid < 32) {  // one wave drives the Tensor Data Mover
    typedef __attribute__((address_space(3))) float* lptr;
    const unsigned lb = (unsigned)(unsigned long long)(lptr)smem;
    tdm_load_2d(bC + b * Kk * Nn + nblk, lb + LDS_BC * 4u, TW, Kk, Nn);
    tdm_load_2d(bS + b * Kk * Nn + nblk, lb + LDS_BS * 4u, TW, Kk, Nn);
    tdm_load_2d(x  + b * Cc * Nn + nblk, lb + LDS_X  * 4u, TW, Cc, Nn);
    tdm_load_2d(h  + b * Cc * Nn + nblk, lb + LDS_H  * 4u, TW, Cc, Nn);
    __builtin_amdgcn_s_wait_tensorcnt(0);
  }
  __syncthreads();

  v8f acc = {0.f, 0.f, 0.f, 0.f, 0.f, 0.f, 0.f, 0.f};
#pragma unroll 4
  for (int k = 0; k < Kk; k += 4) {  // +2*f_c @ cos
    const int kr = k + 2 * half;
    v2f a, bf;
    a.x = fc2[(b * Cc + i0 + l) * Kk + kr];
    a.y = fc2[(b * Cc + i0 + l) * Kk + kr + 1];
    bf.x = smem[LDS_BC + kr * TP + col];
    bf.y = smem[LDS_BC + (kr + 1) * TP + col];
    acc = wmma4(a, bf, acc);
  }
#pragma unroll 4
  for (int k = 0; k < Kk; k += 4) {  // -2*f_s @ sin
    const int kr = k + 2 * half;
    v2f a, bf;
    a.x = fs2[(b * Cc + i0 + l) * Kk + kr];
    a.y = fs2[(b * Cc + i0 + l) * Kk + kr + 1];
    bf.x = smem[LDS_BS + kr * TP + col];
    bf.y = smem[LDS_BS + (kr + 1) * TP + col];
    acc = wmma4(a, bf, acc);
  }
#pragma unroll
  for (int c = 0; c < Cc; c += 4) {  // x2 = W @ x
    const int cr = c + 2 * half;
    v2f a, bf;
    a.x = W[(i0 + l) * Cc + cr];
    a.y = W[(i0 + l) * Cc + cr + 1];
    bf.x = smem[LDS_X + cr * TP + col];
    bf.y = smem[LDS_X + (cr + 1) * TP + col];
    acc = wmma4(a, bf, acc);
  }
#pragma unroll
  for (int c = 0; c < Cc; c += 4) {  // x4 = w2 @ h
    const int cr = c + 2 * half;
    v2f a, bf;
    a.x = w2[(i0 + l) * Cc + cr];
    a.y = w2[(i0 + l) * Cc + cr + 1];
    bf.x = smem[LDS_H + cr * TP + col];
    bf.y = smem[LDS_H + (cr + 1) * TP + col];
    acc = wmma4(a, bf, acc);
  }
#pragma unroll
  for (int j = 0; j < 8; ++j) {
    const int o = i0 + j + 8 * half, n = nblk + col;
    float v = acc[j] + f0[b * Cc + o] + x3[(b * Nn + n) * Cc + o];
    out[(b * Cc + o) * Nn + n] = 0.5f * v * (1.f + erff(v * 0.70710678118654752440f));
  }
}

extern "C" void kernel_launch(void* const* d_in, const int* in_sizes, int n_in,
                              void* d_out, int out_size, void* d_ws, size_t ws_size,
                              hipStream_t stream) {
  (void)in_sizes; (void)n_in; (void)out_size; (void)ws_size;
  const float* x      = (const float*)d_in[0];
  const float* nodes  = (const float*)d_in[1];
  const float* nodew  = (const float*)d_in[2];
  const float* geo    = (const float*)d_in[3];
  const int*   edges  = (const int*)d_in[4];
  const float* egw    = (const float*)d_in[5];
  const float* modes  = (const float*)d_in[6];
  const float* wc     = (const float*)d_in[7];
  const float* wsn    = (const float*)d_in[8];
  const float* w0     = (const float*)d_in[9];
  const float* W      = (const float*)d_in[10];
  const float* gw     = (const float*)d_in[11];
  const float* geo_wx = (const float*)d_in[12];
  const float* wx     = (const float*)d_in[13];
  const float* w2     = (const float*)d_in[14];
  float* ws  = (float*)d_ws;
  float* out = (float*)d_out;

  zero_kernel<<<4096, 256, 0, stream>>>(ws + OFF_XC, ZERO_CNT);
  bases_kernel<<<(Bb * Nn + 255) / 256, 256, 0, stream>>>(nodes, modes,
                                                          ws + OFF_BC, ws + OFF_BS);
  forward_kernel<<<dim3(Nn / 256, Bb), 256, 0, stream>>>(x, nodew, ws + OFF_BC,
      ws + OFF_BS, ws + OFF_XC, ws + OFF_XS, ws + OFF_X0);
  middle_kernel<<<Bb * Kk, Cc, 0, stream>>>(ws + OFF_XC, ws + OFF_XS, ws + OFF_X0,
      wc, wsn, w0, ws + OFF_FC2, ws + OFF_FS2, ws + OFF_F0);
  for (int d = 0; d < Dd; ++d)
    gemm64_kernel<<<dim3(Nn / 32, Bb), 256, 0, stream>>>(gw, Cc * Dd, Dd, d, x,
        ws + OFF_Y + d * (Bb * Nn * Cc), 1);
  gemm64_kernel<<<dim3(Nn / 32, Bb), 256, 0, stream>>>(wx, Cc, 1, 0, x,
      ws + OFF_H, 0);
  h_kernel<<<(Bb * Cc * Nn + 255) / 256, 256, 0, stream>>>(geo_wx, geo, ws + OFF_H);
  edge_kernel<<<(Bb * Ee * 32) / 256, 256, 0, stream>>>(edges, egw, ws + OFF_Y,
                                                        ws + OFF_X3);
  final_kernel<<<dim3(Nn / TW, Bb), 256, 0, stream>>>(ws + OFF_FC2, ws + OFF_FS2,
      ws + OFF_F0, ws + OFF_BC, ws + OFF_BS, W, x, w2, ws + OFF_H, ws + OFF_X3, out);
}